// SCNCoupleDeConvTest_26963804684450
// MI455X (gfx1250) — compile-verified
//
#include <hip/hip_runtime.h>

typedef __attribute__((ext_vector_type(2))) float v2f;
typedef __attribute__((ext_vector_type(8))) float v8f;

#define CIN   16
#define COUT  32
#define DG    128
#define HG    64
#define D3    (DG*DG*DG)           // 2097152
#define HCELLS (HG*HG*HG)          // 262144

// -------------------------------------------------------------------------
// Parity-expanded weights:
//   wbig1[cin*256 + par*32 + co]  = W1[co][cin][par]     (par = fine-voxel parity zyx)
//   w2big[co*128  + par*16 + ci]  = W2[ci][co][7 ^ par]  (deconv kernel idx = ~parity)
// -------------------------------------------------------------------------
__global__ void prep_weights_kernel(const float* __restrict__ W1,
                                    const float* __restrict__ W2,
                                    float* __restrict__ wbig1,
                                    float* __restrict__ w2big) {
  int i = blockIdx.x * blockDim.x + threadIdx.x;
  if (i < 16 * 256) {
    int cin = i >> 8, par = (i >> 5) & 7, co = i & 31;
    wbig1[i] = W1[(co * CIN + cin) * 8 + par];
    int co2 = i >> 7, par2 = (i >> 4) & 7, ci = i & 15;
    w2big[i] = W2[(ci * COUT + co2) * 8 + (7 ^ par2)];
  }
}

// -------------------------------------------------------------------------
// Scatter: per 16-point tile (one wave), P(16x256) = F(16x16) @ Wbig1(16x256)
// via v_wmma_f32_16x16x4_f32; each point's matching 32-wide parity block is
// atomically added into h[b][cell][co]. N-tiles whose parity is absent in
// the wave are skipped entirely (uniform branch, EXEC stays all-1).
// -------------------------------------------------------------------------
__global__ void scn_scatter_kernel(const float* __restrict__ feats,
                                   const int*   __restrict__ coors,
                                   const float* __restrict__ wbig1,
                                   float*       __restrict__ h,
                                   int n) {
  __shared__ float    lwb[16 * 256];         // 16 KB Wbig1 copy
  __shared__ unsigned s_hbase[8 * 16];
  __shared__ int      s_par[8 * 16];
  __shared__ unsigned s_pmask[8];

  for (int i = threadIdx.x; i < 16 * 256; i += blockDim.x) lwb[i] = wbig1[i];

  const int wave  = threadIdx.x >> 5;
  const int lane  = threadIdx.x & 31;
  const int tile0 = (blockIdx.x * 8 + wave) * 16;

  if (lane < 16) {
    if (lane == 0) s_pmask[wave] = 0u;       // in-order LDS ops within wave
    int p = tile0 + lane;
    int valid = (p < n);
    int pc = valid ? p : (n - 1);
    int bb = coors[pc * 4 + 0];
    int z  = coors[pc * 4 + 1];
    int y  = coors[pc * 4 + 2];
    int x  = coors[pc * 4 + 3];
    int cell = ((z >> 1) * HG + (y >> 1)) * HG + (x >> 1);
    int par  = ((z & 1) << 2) | ((y & 1) << 1) | (x & 1);
    s_hbase[wave * 16 + lane] = ((unsigned)bb * HCELLS + (unsigned)cell) * 32u;
    s_par  [wave * 16 + lane] = valid ? par : 255;   // sentinel never matches
    if (valid) atomicOr(&s_pmask[wave], 1u << par);
  }
  __syncthreads();

  // A fragments: F tile (16 pts x 16 cin), K-chunks of 4.
  // Lanes 0-15: K = 4kk+{0,1};  lanes 16-31: K = 4kk+{2,3}.
  const int row0 = (lane >> 4) << 1;   // 0 or 2
  const int ncol = lane & 15;
  v2f a[4];
  {
    int p = tile0 + (lane & 15);
    if (p >= n) p = n - 1;
    const float* fp = feats + (size_t)p * CIN + row0;
#pragma unroll
    for (int kk = 0; kk < 4; ++kk) {
      a[kk].x = fp[4 * kk + 0];
      a[kk].y = fp[4 * kk + 1];
    }
  }

  // Hoist per-lane point metadata (loop-invariant across t) into registers.
  const int mbase = (lane >> 4) << 3;  // C rows: lanes16-31 hold M = r+8
  unsigned hb8[8];
  int      pr8[8];
#pragma unroll
  for (int r = 0; r < 8; ++r) {
    int idx = wave * 16 + mbase + r;
    hb8[r] = s_hbase[idx];
    pr8[r] = s_par[idx];
  }
  const unsigned pmask = s_pmask[wave];

  for (int t = 0; t < 16; ++t) {
    const int par = t >> 1;                      // uniform over the N-tile
    if (!((pmask >> par) & 1u)) continue;        // uniform skip: no such parity
    v8f c = {0.f, 0.f, 0.f, 0.f, 0.f, 0.f, 0.f, 0.f};
#pragma unroll
    for (int kk = 0; kk < 4; ++kk) {
      v2f bf;
      bf.x = lwb[(4 * kk + row0    ) * 256 + t * 16 + ncol];
      bf.y = lwb[(4 * kk + row0 + 1) * 256 + t * 16 + ncol];
      c = __builtin_amdgcn_wmma_f32_16x16x4_f32(false, a[kk], false, bf,
                                                (short)0, c, false, false);
    }
    const int co = ((t & 1) << 4) | ncol;        // col & 31
#pragma unroll
    for (int r = 0; r < 8; ++r) {
      if (pr8[r] == par)
        atomicAdd(&h[hb8[r] + (unsigned)co], c[r]);
    }
  }
}

// -------------------------------------------------------------------------
// Gather: per 16-point tile, Dec(16x128) = H(16x32) @ W2big(32x128) via WMMA;
// each point's matching 16-wide parity block is stored to the dense output.
// -------------------------------------------------------------------------
__global__ void scn_gather_kernel(const float* __restrict__ h,
                                  const int*   __restrict__ coors,
                                  const float* __restrict__ w2big,
                                  float*       __restrict__ out,
                                  int n) {
  __shared__ float    lwb[32 * 128];         // 16 KB W2big copy
  __shared__ unsigned s_hbase[8 * 16];
  __shared__ unsigned s_obase[8 * 16];
  __shared__ int      s_par[8 * 16];
  __shared__ unsigned s_pmask[8];

  for (int i = threadIdx.x; i < 32 * 128; i += blockDim.x) lwb[i] = w2big[i];

  const int wave  = threadIdx.x >> 5;
  const int lane  = threadIdx.x & 31;
  const int tile0 = (blockIdx.x * 8 + wave) * 16;

  if (lane < 16) {
    if (lane == 0) s_pmask[wave] = 0u;
    int p = tile0 + lane;
    int valid = (p < n);
    int pc = valid ? p : (n - 1);
    int bb = coors[pc * 4 + 0];
    int z  = coors[pc * 4 + 1];
    int y  = coors[pc * 4 + 2];
    int x  = coors[pc * 4 + 3];
    int cell = ((z >> 1) * HG + (y >> 1)) * HG + (x >> 1);
    int par  = ((z & 1) << 2) | ((y & 1) << 1) | (x & 1);
    s_hbase[wave * 16 + lane] = ((unsigned)bb * HCELLS + (unsigned)cell) * 32u;
    s_obase[wave * 16 + lane] = (unsigned)bb * (CIN * D3) +
                                (unsigned)((z * DG + y) * DG + x);
    s_par  [wave * 16 + lane] = valid ? par : 255;
    if (valid) atomicOr(&s_pmask[wave], 1u << par);
  }
  __syncthreads();

  const int row0 = (lane >> 4) << 1;
  const int ncol = lane & 15;

  // A fragments: H rows (16 pts x 32 co), 8 K-chunks of 4.
  v2f a[8];
  {
    const float* hp = h + s_hbase[wave * 16 + (lane & 15)] + row0;
#pragma unroll
    for (int kk = 0; kk < 8; ++kk) {
      a[kk].x = hp[4 * kk + 0];
      a[kk].y = hp[4 * kk + 1];
    }
  }

  const int mbase = (lane >> 4) << 3;
  unsigned ob8[8];
  int      pr8[8];
#pragma unroll
  for (int r = 0; r < 8; ++r) {
    int idx = wave * 16 + mbase + r;
    ob8[r] = s_obase[idx];
    pr8[r] = s_par[idx];
  }
  const unsigned pmask = s_pmask[wave];

  for (int t = 0; t < 8; ++t) {
    if (!((pmask >> t) & 1u)) continue;          // par == t, uniform skip
    v8f c = {0.f, 0.f, 0.f, 0.f, 0.f, 0.f, 0.f, 0.f};
#pragma unroll
    for (int kk = 0; kk < 8; ++kk) {
      v2f bf;
      bf.x = lwb[(4 * kk + row0    ) * 128 + t * 16 + ncol];
      bf.y = lwb[(4 * kk + row0 + 1) * 128 + t * 16 + ncol];
      c = __builtin_amdgcn_wmma_f32_16x16x4_f32(false, a[kk], false, bf,
                                                (short)0, c, false, false);
    }
#pragma unroll
    for (int r = 0; r < 8; ++r) {
      if (pr8[r] == t)
        out[ob8[r] + (unsigned)ncol * D3] = c[r];   // ci = col & 15 = ncol
    }
  }
}

// -------------------------------------------------------------------------
extern "C" void kernel_launch(void* const* d_in, const int* in_sizes, int n_in,
                              void* d_out, int out_size, void* d_ws, size_t ws_size,
                              hipStream_t stream) {
  const float* feats = (const float*)d_in[0];
  const float* W1    = (const float*)d_in[1];
  const float* W2    = (const float*)d_in[2];
  const int*   coors = (const int*)d_in[3];
  float*       out   = (float*)d_out;

  const int n   = in_sizes[0] / CIN;          // number of points
  const int Bsz = out_size / (CIN * D3);      // batch size

  float* wbig1 = (float*)d_ws;                            // 16 KB
  float* w2big = wbig1 + 16 * 256;                        // 16 KB
  float* hbuf  = (float*)((char*)d_ws + 32 * 1024);       // B * 64^3 * 32 floats
  const size_t hbytes = (size_t)Bsz * HCELLS * 32 * sizeof(float);

  (void)hipMemsetAsync(out, 0, (size_t)out_size * sizeof(float), stream);
  (void)hipMemsetAsync(hbuf, 0, hbytes, stream);

  prep_weights_kernel<<<16, 256, 0, stream>>>(W1, W2, wbig1, w2big);

  const int tiles  = (n + 15) / 16;           // one wave per 16-point tile
  const int blocks = (tiles + 7) / 8;         // 8 waves per 256-thread block
  scn_scatter_kernel<<<blocks, 256, 0, stream>>>(feats, coors, wbig1, hbuf, n);
  scn_gather_kernel <<<blocks, 256, 0, stream>>>(hbuf, coors, w2big, out, n);
}